// GCN_8589935121
// MI455X (gfx1250) — compile-verified
//
#include <hip/hip_runtime.h>
#include <stdint.h>

// ---------------------------------------------------------------------------
// 2-layer GCN (DGL GraphConv, norm='both') for MI455X / gfx1250.
// Memory-bound: ~2.5 GB edge gather/scatter traffic dominates; GEMMs (~5
// GFLOP) are done with V_WMMA_F32_16X16X4_F32 (full fp32 precision, since the
// matrix units are nowhere near the bottleneck).
// ---------------------------------------------------------------------------

#define IN_DIM 64
#define HID    128

typedef __attribute__((ext_vector_type(2))) float v2f;
typedef __attribute__((ext_vector_type(8))) float v8f;

// ---------------- zero fill (accumulators + degree counters) ----------------
__global__ void gcn_fill0(float* __restrict__ p, long long n) {
  long long i = (long long)blockIdx.x * blockDim.x + threadIdx.x;
  if (i < n) p[i] = 0.0f;
}

// ---------------- degree counting via f32 atomics ---------------------------
__global__ void gcn_degree(const int* __restrict__ src, const int* __restrict__ dst,
                           float* __restrict__ dego, float* __restrict__ degi, int e) {
  int i = blockIdx.x * blockDim.x + threadIdx.x;
  if (i < e) {
    atomicAdd(&dego[src[i]], 1.0f);
    atomicAdd(&degi[dst[i]], 1.0f);
  }
}

// ---------------- deg -> rsqrt(max(deg,1)) in place -------------------------
__global__ void gcn_rsqrt(float* __restrict__ d, int n) {
  int i = blockIdx.x * blockDim.x + threadIdx.x;
  if (i < n) {
    float v = d[i];
    v = (v < 1.0f) ? 1.0f : v;
    d[i] = rsqrtf(v);
  }
}

// ---------------- row-scale: out[r,:] = x[r,:] * s[r] (float4) --------------
__global__ void gcn_scale_rows(const float* __restrict__ x, const float* __restrict__ s,
                               float* __restrict__ out, long long n4, int f4) {
  long long i = (long long)blockIdx.x * blockDim.x + threadIdx.x;
  if (i >= n4) return;
  int row = (int)(i / f4);
  float4 v = ((const float4*)x)[i];
  float sc = s[row];
  float4 o;
  o.x = v.x * sc; o.y = v.y * sc; o.z = v.z * sc; o.w = v.w * sc;
  ((float4*)out)[i] = o;
}

// ---------------- edge scatter: agg[dst] += msg[src] (per 4-float chunk) ----
__global__ void gcn_scatter(const float* __restrict__ msg, const int* __restrict__ src,
                            const int* __restrict__ dst, float* __restrict__ agg,
                            long long total, int chunks, int F) {
  long long t = (long long)blockIdx.x * blockDim.x + threadIdx.x;
  if (t >= total) return;
  int e = (int)(t / chunks);
  int c = (int)(t % chunks);
  int s = src[e];
  int d = dst[e];
  const float4 v = *(const float4*)(msg + (long long)s * F + c * 4);
  float* o = agg + (long long)d * F + c * 4;
  atomicAdd(o + 0, v.x);
  atomicAdd(o + 1, v.y);
  atomicAdd(o + 2, v.z);
  atomicAdd(o + 3, v.w);
}

// ---------------- fused: out = act((A*rs) @ W + b) * (POST ? ps : 1) --------
// A: [n,K] fp32, W: [K,128], 16 rows per block, 8 waves x 16-col tiles.
// n % 16 == 0 and K % 4 == 0 are required (true here: n=100000, K=64/128),
// so EXEC is all-ones for every WMMA (ISA requirement).
template <int K, bool LEAKY, bool POST>
__global__ __launch_bounds__(256) void gcn_gemm_wmma(
    const float* __restrict__ A,   // aggregated features [n, K]
    const float* __restrict__ rs,  // in_inv [n]  (pre-GEMM row scale)
    const float* __restrict__ W,   // [K, 128]
    const float* __restrict__ bias,// [128]
    const float* __restrict__ ps,  // out_inv [n] (post scale) or nullptr
    float* __restrict__ out,       // [n, 128]
    int n) {
  constexpr int C   = 128;
  constexpr int LDA = K + 4;                 // pad to avoid LDS bank conflicts
  __shared__ __align__(16) float As[16 * LDA];

  const int m0  = blockIdx.x * 16;
  const int tid = threadIdx.x;

  // Cooperatively stage the 16xK A tile, scaled by in_inv[row].
  constexpr int TOT4 = 16 * K / 4;           // float4 elements in the tile
  #pragma unroll
  for (int i4 = tid; i4 < TOT4; i4 += 256) {
    int idx = i4 * 4;
    int r = idx / K;
    int c = idx % K;
    float4 v = *(const float4*)(A + (long long)(m0 + r) * K + c);
    float sc = rs[m0 + r];
    float* dstp = &As[r * LDA + c];
    dstp[0] = v.x * sc;
    dstp[1] = v.y * sc;
    dstp[2] = v.z * sc;
    dstp[3] = v.w * sc;
  }
  __syncthreads();

  const int wave = tid >> 5;                 // 0..7 -> 16-col slab
  const int lane = tid & 31;
  const int hl   = lane >> 4;                // half-wave select (K sub-pair)
  const int lr   = lane & 15;                // A row / B col within tile
  const int n0   = wave * 16;

  v8f acc = {};
  #pragma unroll
  for (int k = 0; k < K; k += 4) {
    const int ka = k + 2 * hl;               // lanes 0-15: K=k,k+1 ; 16-31: k+2,k+3
    v2f a;
    a[0] = As[lr * LDA + ka];
    a[1] = As[lr * LDA + ka + 1];
    v2f b;
    b[0] = W[ka * C + n0 + lr];
    b[1] = W[(ka + 1) * C + n0 + lr];
    acc = __builtin_amdgcn_wmma_f32_16x16x4_f32(
        /*neg_a=*/false, a, /*neg_b=*/false, b,
        /*c_mod=*/(short)0, acc, /*reuse_a=*/false, /*reuse_b=*/false);
  }

  // Epilogue: D layout -> lane(lr) holds col n0+lr; VGPR v holds row m0+v+8*hl.
  const float bv = bias[n0 + lr];
  #pragma unroll
  for (int v = 0; v < 8; ++v) {
    const int row = m0 + v + 8 * hl;
    float val = acc[v] + bv;
    if (LEAKY) val = (val > 0.0f) ? val : 0.01f * val;
    if (POST)  val *= ps[row];
    out[(long long)row * C + n0 + lr] = val;
  }
}

// ---------------------------------------------------------------------------
extern "C" void kernel_launch(void* const* d_in, const int* in_sizes, int n_in,
                              void* d_out, int out_size, void* d_ws, size_t ws_size,
                              hipStream_t stream) {
  const float* x  = (const float*)d_in[0];  // [n, 64]
  const int*  src = (const int*)d_in[1];    // [e]
  const int*  dst = (const int*)d_in[2];    // [e]
  const float* W1 = (const float*)d_in[3];  // [64, 128]
  const float* b1 = (const float*)d_in[4];  // [128]
  const float* W2 = (const float*)d_in[5];  // [128, 128]
  const float* b2 = (const float*)d_in[6];  // [128]
  float* out = (float*)d_out;               // [n, 128]

  const int n = in_sizes[0] / IN_DIM;       // 100000 (multiple of 16)
  const int e = in_sizes[1];                // 1600000

  // Workspace layout (fp32). Zeroed region is contiguous and first.
  float* ws   = (float*)d_ws;
  float* agg1 = ws;                                   // n*64   (zeroed)
  float* agg2 = agg1 + (long long)n * IN_DIM;         // n*128  (zeroed)
  float* dego = agg2 + (long long)n * HID;            // n      (zeroed -> out_inv)
  float* degi = dego + n;                             // n      (zeroed -> in_inv)
  float* xs   = degi + n;                             // n*64   (x * out_inv)
  float* h1s  = xs + (long long)n * IN_DIM;           // n*128  (h1 * out_inv)

  // 1. zero accumulators + degree counters (fresh every call)
  const long long nz = (long long)n * (IN_DIM + HID + 2);
  gcn_fill0<<<(unsigned)((nz + 255) / 256), 256, 0, stream>>>(agg1, nz);

  // 2. degrees, 3. rsqrt(max(deg,1)) (dego/degi are contiguous -> one pass)
  gcn_degree<<<(e + 255) / 256, 256, 0, stream>>>(src, dst, dego, degi, e);
  gcn_rsqrt<<<(2 * n + 255) / 256, 256, 0, stream>>>(dego, 2 * n);

  // 4. xs = x * out_inv
  const long long x4 = (long long)n * IN_DIM / 4;
  gcn_scale_rows<<<(unsigned)((x4 + 255) / 256), 256, 0, stream>>>(x, dego, xs, x4, IN_DIM / 4);

  // 5. layer-1 edge scatter: agg1[dst] += xs[src]
  const long long t1 = (long long)e * (IN_DIM / 4);
  gcn_scatter<<<(unsigned)((t1 + 255) / 256), 256, 0, stream>>>(xs, src, dst, agg1, t1,
                                                                IN_DIM / 4, IN_DIM);

  // 6. h1s = leaky_relu((agg1*in_inv) @ W1 + b1) * out_inv   (WMMA fp32)
  gcn_gemm_wmma<IN_DIM, true, true><<<n / 16, 256, 0, stream>>>(agg1, degi, W1, b1,
                                                                dego, h1s, n);

  // 7. layer-2 edge scatter: agg2[dst] += h1s[src]
  const long long t2 = (long long)e * (HID / 4);
  gcn_scatter<<<(unsigned)((t2 + 255) / 256), 256, 0, stream>>>(h1s, src, dst, agg2, t2,
                                                                HID / 4, HID);

  // 8. out = (agg2*in_inv) @ W2 + b2   (WMMA fp32)
  gcn_gemm_wmma<HID, false, false><<<n / 16, 256, 0, stream>>>(agg2, degi, W2, b2,
                                                               nullptr, out, n);
}